// SGCNodeCLF_16020228014933
// MI455X (gfx1250) — compile-verified
//
#include <hip/hip_runtime.h>
#include <hip/hip_bf16.h>
#include <math.h>

typedef __attribute__((ext_vector_type(2))) float v2f;
typedef __attribute__((ext_vector_type(8))) float v8f;

#define D_DIM 128
#define C_DIM 40

// ---------------------------------------------------------------------------
// Degree computation: deg[i] = 1 (self loop) + #incoming edges, then rsqrt.
// ---------------------------------------------------------------------------
__global__ void k_deg_init(float* __restrict__ deg, int n) {
    int i = blockIdx.x * blockDim.x + threadIdx.x;
    if (i < n) deg[i] = 1.0f;
}

__global__ void k_deg_edges(const int* __restrict__ col, float* __restrict__ deg, int e) {
    int i = blockIdx.x * blockDim.x + threadIdx.x;
    if (i < e) unsafeAtomicAdd(&deg[col[i]], 1.0f);  // global_atomic_add_f32
}

__global__ void k_rsqrt(float* __restrict__ deg, int n) {
    int i = blockIdx.x * blockDim.x + threadIdx.x;
    if (i < n) deg[i] = rsqrtf(deg[i]);   // deg >= 1 always (self loop)
}

// ---------------------------------------------------------------------------
// One propagation hop, part 1: self-loop term, also initializes hdst fully.
// hdst[i,:] = dis[i]^2 * hsrc[i,:]    (32 lanes x float4 = 128 floats/node)
// ---------------------------------------------------------------------------
__global__ void k_self(const float* __restrict__ hsrc, float* __restrict__ hdst,
                       const float* __restrict__ dis, int n) {
    int t    = blockIdx.x * blockDim.x + threadIdx.x;
    int node = t >> 5;
    int q    = (t & 31) * 4;
    if (node >= n) return;
    float s = dis[node];
    s = s * s;
    float4 v = *(const float4*)(hsrc + (size_t)node * D_DIM + q);
    float4 o; o.x = v.x * s; o.y = v.y * s; o.z = v.z * s; o.w = v.w * s;
    *(float4*)(hdst + (size_t)node * D_DIM + q) = o;
}

// ---------------------------------------------------------------------------
// One propagation hop, part 2: edge scatter. One wave32 per edge; lane l
// handles 4 consecutive floats. Coalesced 512B read of h[row], 4 hardware
// f32 atomic adds into h[col]. Edge metadata is wave-uniform (L1 broadcast).
// ---------------------------------------------------------------------------
__global__ void k_scatter(const float* __restrict__ hsrc, float* __restrict__ hdst,
                          const int* __restrict__ row, const int* __restrict__ col,
                          const float* __restrict__ dis, int e) {
    int gt   = blockIdx.x * blockDim.x + threadIdx.x;
    int edge = gt >> 5;
    int lane = gt & 31;
    if (edge >= e) return;
    int   r = row[edge];
    int   c = col[edge];
    float w = dis[r] * dis[c];
    float4 v = *(const float4*)(hsrc + (size_t)r * D_DIM + lane * 4);
    float* dp = hdst + (size_t)c * D_DIM + lane * 4;
    unsafeAtomicAdd(dp + 0, v.x * w);
    unsafeAtomicAdd(dp + 1, v.y * w);
    unsafeAtomicAdd(dp + 2, v.z * w);
    unsafeAtomicAdd(dp + 3, v.w * w);
}

// ---------------------------------------------------------------------------
// Dense layer with f32 WMMA: logits[16 rows x 40 cols] per wave.
//   A (16x4 f32): lane<16 holds row m=lane, k = kb + {0,1};
//                 lane>=16 same rows,       k = kb + {2,3}.
//   B (4x16 f32): lane&15 = output col n, k = vgpr + 2*(lane>=16);
//                 B(k, c0+n) = W[c0+n][kb+k]  (W is [C, D] row-major).
//   C/D (16x16):  lane&15 = n, vgpr r -> m = r + 8*(lane>=16).
// Three column tiles (c0 = 0,16,32) share the A operand; tile 2 masked to
// the 8 valid columns. One wave per block -> wave-uniform control flow,
// EXEC is all ones at every WMMA.
// ---------------------------------------------------------------------------
__global__ __launch_bounds__(32)
void k_linear_wmma(const float* __restrict__ h, const float* __restrict__ W,
                   const float* __restrict__ bias, float* __restrict__ out) {
    const int rb   = blockIdx.x;        // 16-row block index
    const int lane = threadIdx.x;       // 0..31
    const int nn   = lane & 15;
    const int half = lane >> 4;         // 0 or 1
    const int row0 = rb * 16;

    const float* arow = h + (size_t)(row0 + nn) * D_DIM + 2 * half;

    const int   c2    = 32 + nn;
    const bool  v2ok  = (c2 < C_DIM);
    const float mask2 = v2ok ? 1.0f : 0.0f;
    const float* wp0 = W + (size_t)(0  + nn) * D_DIM + 2 * half;
    const float* wp1 = W + (size_t)(16 + nn) * D_DIM + 2 * half;
    const float* wp2 = W + (size_t)(v2ok ? c2 : 0) * D_DIM + 2 * half;

    v8f acc0 = {}, acc1 = {}, acc2 = {};

#pragma unroll 4
    for (int kb = 0; kb < D_DIM; kb += 4) {
        v2f a, b0, b1, b2;
        a.x  = arow[kb];     a.y  = arow[kb + 1];
        b0.x = wp0[kb];      b0.y = wp0[kb + 1];
        b1.x = wp1[kb];      b1.y = wp1[kb + 1];
        b2.x = wp2[kb] * mask2;
        b2.y = wp2[kb + 1] * mask2;
        acc0 = __builtin_amdgcn_wmma_f32_16x16x4_f32(false, a, false, b0, (short)0, acc0, false, false);
        acc1 = __builtin_amdgcn_wmma_f32_16x16x4_f32(false, a, false, b1, (short)0, acc1, false, false);
        acc2 = __builtin_amdgcn_wmma_f32_16x16x4_f32(false, a, false, b2, (short)0, acc2, false, false);
    }

    const float bia0 = bias[nn];
    const float bia1 = bias[16 + nn];
    const float bia2 = v2ok ? bias[c2] : 0.0f;

#pragma unroll
    for (int r = 0; r < 8; ++r) {
        int m = r + 8 * half;
        float* orow = out + (size_t)(row0 + m) * C_DIM;
        orow[nn]      = acc0[r] + bia0;
        orow[16 + nn] = acc1[r] + bia1;
        if (v2ok) orow[c2] = acc2[r] + bia2;
    }
}

// ---------------------------------------------------------------------------
// In-place row-wise log_softmax over C_DIM classes. One thread per row.
// ---------------------------------------------------------------------------
__global__ void k_logsoftmax(float* __restrict__ out, int n) {
    int i = blockIdx.x * blockDim.x + threadIdx.x;
    if (i >= n) return;
    float* p = out + (size_t)i * C_DIM;
    float m = -INFINITY;
#pragma unroll
    for (int c = 0; c < C_DIM; ++c) m = fmaxf(m, p[c]);
    float s = 0.0f;
#pragma unroll
    for (int c = 0; c < C_DIM; ++c) s += __expf(p[c] - m);
    float lse = m + __logf(s);
#pragma unroll
    for (int c = 0; c < C_DIM; ++c) p[c] = p[c] - lse;
}

// ---------------------------------------------------------------------------
extern "C" void kernel_launch(void* const* d_in, const int* in_sizes, int n_in,
                              void* d_out, int out_size, void* d_ws, size_t ws_size,
                              hipStream_t stream) {
    const float* x  = (const float*)d_in[0];
    const int*   ei = (const int*)  d_in[1];
    const float* W  = (const float*)d_in[2];
    const float* b  = (const float*)d_in[3];
    float*       out = (float*)d_out;

    const int N = in_sizes[0] / D_DIM;   // 50000
    const int E = in_sizes[1] / 2;       // 800000
    const int* row = ei;
    const int* col = ei + E;

    // Workspace layout: hA | hB | dis  (all float32)
    float* hA  = (float*)d_ws;
    float* hB  = hA + (size_t)N * D_DIM;
    float* dis = hB + (size_t)N * D_DIM;

    const int TB = 256;

    // --- degrees & normalization ---
    k_deg_init <<<(N + TB - 1) / TB, TB, 0, stream>>>(dis, N);
    k_deg_edges<<<(E + TB - 1) / TB, TB, 0, stream>>>(col, dis, E);
    k_rsqrt    <<<(N + TB - 1) / TB, TB, 0, stream>>>(dis, N);

    const int selfBlocks = (int)(((size_t)N * 32 + TB - 1) / TB);
    const int scatBlocks = (int)(((size_t)E * 32 + TB - 1) / TB);

    // --- hop 1: x -> hA ---
    k_self   <<<selfBlocks, TB, 0, stream>>>(x, hA, dis, N);
    k_scatter<<<scatBlocks, TB, 0, stream>>>(x, hA, row, col, dis, E);
    // --- hop 2: hA -> hB ---
    k_self   <<<selfBlocks, TB, 0, stream>>>(hA, hB, dis, N);
    k_scatter<<<scatBlocks, TB, 0, stream>>>(hA, hB, row, col, dis, E);
    // --- hop 3: hB -> hA ---
    k_self   <<<selfBlocks, TB, 0, stream>>>(hB, hA, dis, N);
    k_scatter<<<scatBlocks, TB, 0, stream>>>(hB, hA, row, col, dis, E);

    // --- dense layer via f32 WMMA (N divisible by 16: 50000 = 3125*16) ---
    k_linear_wmma<<<N / 16, 32, 0, stream>>>(hA, W, b, out);

    // --- log_softmax in place on d_out ---
    k_logsoftmax<<<(N + TB - 1) / TB, TB, 0, stream>>>(out, N);
}